// ConvexSoftMixer_26628797235346
// MI455X (gfx1250) — compile-verified
//
#include <hip/hip_runtime.h>
#include <hip/hip_bf16.h>

typedef __attribute__((ext_vector_type(2))) float v2f;
typedef __attribute__((ext_vector_type(8))) float v8f;

#if __has_builtin(__builtin_amdgcn_global_load_async_to_lds_b32) && \
    __has_builtin(__builtin_amdgcn_s_wait_asynccnt)
#define USE_ASYNC_LDS 1
#else
#define USE_ASYNC_LDS 0
#endif

typedef __attribute__((address_space(1))) int g_int;
typedef __attribute__((address_space(3))) int l_int;

// D = A(16x4) * B(4x16) + C, full fp32 WMMA on gfx1250.
// Per-lane: A = {A[m][k0], A[m][k0+1]}, B = {B[k0][n], B[k0+1][n]} with
// m = n = lane%16, k0 = 2*(lane/16).  For x@W^T both operands load row
// (lane%16) of a row-major matrix at columns k0..k0+1.
__device__ __forceinline__ v8f wmma4(v2f a, v2f b, v8f c) {
  return __builtin_amdgcn_wmma_f32_16x16x4_f32(false, a, false, b, (short)0, c,
                                               false, false);
}

__device__ __forceinline__ float softplusf(float x) {
  // jax.nn.softplus == max(x,0) + log1p(exp(-|x|))
  return fmaxf(x, 0.0f) + log1pf(expf(-fabsf(x)));
}

// ---------------------------------------------------------------------------
// Kernel 1: per-token work for all B*H*S = 4096 tokens.
//   f_q = sum softplus(softplus(q@W1q^T+b1)@W2q^T+b2)   (same for k -> g_k)
//   phi_q = exp(q@Wh^T) [tok][p], phi_kT = exp(k@Wh^T) stored [h][p][t]
//   u = v@Wv^T [tok][p]
// Block = 128 threads (4 waves); each wave owns a 16-token tile.
// ---------------------------------------------------------------------------
__global__ __launch_bounds__(128) void cs_token_kernel(
    const float* __restrict__ q, const float* __restrict__ k,
    const float* __restrict__ v, const float* __restrict__ sq_raw1,
    const float* __restrict__ sq_b1, const float* __restrict__ sq_raw2,
    const float* __restrict__ sq_b2, const float* __restrict__ sk_raw1,
    const float* __restrict__ sk_b1, const float* __restrict__ sk_raw2,
    const float* __restrict__ sk_b2, const float* __restrict__ Wh,
    const float* __restrict__ Wv, float* __restrict__ phi_q,
    float* __restrict__ phi_kT, float* __restrict__ u_ws,
    float* __restrict__ f_q, float* __restrict__ g_k) {
  __shared__ float Wbuf[64 * 66];    // softplus(weights), reused per phase
  __shared__ float Whs[32 * 66];
  __shared__ float Wvs[32 * 66];
  __shared__ float bias_s[64];
  __shared__ float h1s[4][16 * 66];  // per-wave hidden activations
  __shared__ float red[4][32][8];    // row-sum reduction scratch

  const int tid = threadIdx.x;
  const int wave = tid >> 5;
  const int lane = tid & 31;
  const int lh = lane & 15;  // row index of A / B' fragment
  const int kh = lane >> 4;  // which K-half this lane holds
  const int tokBase = blockIdx.x * 64 + wave * 16;

  // Warm the L2/L0 for the raw ICNN weights we will transform shortly
  // (emits global_prefetch_b8; one 128B line per thread per matrix).
  {
    __builtin_prefetch(sq_raw1 + tid * 32, 0, 1);
    __builtin_prefetch(sq_raw2 + tid * 32, 0, 1);
    __builtin_prefetch(sk_raw1 + tid * 32, 0, 1);
    __builtin_prefetch(sk_raw2 + tid * 32, 0, 1);
  }

  // Wh / Wv go to LDS verbatim -> use the gfx1250 async global->LDS path
  // (ASYNCcnt-tracked, no VGPR round trip) when the toolchain exposes it.
#if USE_ASYNC_LDS
  for (int i = tid; i < 32 * 64; i += 128) {
    __builtin_amdgcn_global_load_async_to_lds_b32(
        (g_int*)(Wh + i), (l_int*)&Whs[(i >> 6) * 66 + (i & 63)], 0, 0);
    __builtin_amdgcn_global_load_async_to_lds_b32(
        (g_int*)(Wv + i), (l_int*)&Wvs[(i >> 6) * 66 + (i & 63)], 0, 0);
  }
  __builtin_amdgcn_s_wait_asynccnt(0);
#else
  for (int i = tid; i < 32 * 64; i += 128) {
    Whs[(i >> 6) * 66 + (i & 63)] = Wh[i];
    Wvs[(i >> 6) * 66 + (i & 63)] = Wv[i];
  }
#endif
  __syncthreads();

  // ---- phi_q / phi_k / u projections (P = 32 -> two 16-col tiles) ----
  for (int nt = 0; nt < 2; ++nt) {
    v8f aq = {}, ak = {}, av = {};
    for (int kk = 0; kk < 16; ++kk) {
      int k0 = kk * 4 + 2 * kh;
      v2f bq = *(const v2f*)&Whs[(nt * 16 + lh) * 66 + k0];
      v2f bv = *(const v2f*)&Wvs[(nt * 16 + lh) * 66 + k0];
      v2f a1 = *(const v2f*)&q[(tokBase + lh) * 64 + k0];
      v2f a2 = *(const v2f*)&k[(tokBase + lh) * 64 + k0];
      v2f a3 = *(const v2f*)&v[(tokBase + lh) * 64 + k0];
      aq = wmma4(a1, bq, aq);
      ak = wmma4(a2, bq, ak);
      av = wmma4(a3, bv, av);
    }
    for (int j = 0; j < 8; ++j) {
      int tok = tokBase + j + 8 * kh;
      int p = nt * 16 + lh;
      phi_q[tok * 32 + p] = expf(aq[j]);
      int hh = tok >> 9, tt = tok & 511;
      phi_kT[(hh * 32 + p) * 512 + tt] = expf(ak[j]);  // transposed store
      u_ws[tok * 32 + p] = av[j];
    }
  }

  // ---- ICNN scores: which==0 -> f_q(q), which==1 -> g_k(k) ----
  for (int which = 0; which < 2; ++which) {
    const float* x = which ? k : q;
    const float* r1 = which ? sk_raw1 : sq_raw1;
    const float* b1 = which ? sk_b1 : sq_b1;
    const float* r2 = which ? sk_raw2 : sq_raw2;
    const float* b2 = which ? sk_b2 : sq_b2;
    float* outv = which ? g_k : f_q;

    __syncthreads();
    for (int i = tid; i < 64 * 64; i += 128)
      Wbuf[(i >> 6) * 66 + (i & 63)] = softplusf(r1[i]);
    if (tid < 64) bias_s[tid] = b1[tid];
    __syncthreads();

    // layer 1: h1 = softplus(x @ W1^T + b1), 16x64 per wave
    for (int nt = 0; nt < 4; ++nt) {
      v8f acc = {};
      for (int kk = 0; kk < 16; ++kk) {
        int k0 = kk * 4 + 2 * kh;
        v2f a = *(const v2f*)&x[(tokBase + lh) * 64 + k0];
        v2f b = *(const v2f*)&Wbuf[(nt * 16 + lh) * 66 + k0];
        acc = wmma4(a, b, acc);
      }
      int col = nt * 16 + lh;
      for (int j = 0; j < 8; ++j)
        h1s[wave][(j + 8 * kh) * 66 + col] = softplusf(acc[j] + bias_s[col]);
    }
    __syncthreads();
    for (int i = tid; i < 64 * 64; i += 128)
      Wbuf[(i >> 6) * 66 + (i & 63)] = softplusf(r2[i]);
    if (tid < 64) bias_s[tid] = b2[tid];
    __syncthreads();

    // layer 2 + softplus + row-sum over the 64 output features
    float racc[8] = {0, 0, 0, 0, 0, 0, 0, 0};
    for (int nt = 0; nt < 4; ++nt) {
      v8f acc = {};
      for (int kk = 0; kk < 16; ++kk) {
        int k0 = kk * 4 + 2 * kh;
        v2f a = *(const v2f*)&h1s[wave][lh * 66 + k0];
        v2f b = *(const v2f*)&Wbuf[(nt * 16 + lh) * 66 + k0];
        acc = wmma4(a, b, acc);
      }
      float bb = bias_s[nt * 16 + lh];
      for (int j = 0; j < 8; ++j) racc[j] += softplusf(acc[j] + bb);
    }
    for (int j = 0; j < 8; ++j) red[wave][lane][j] = racc[j];
    __syncthreads();
    if (lane < 16) {
      // row m==lane: rows 0-7 live in lanes 0-15 (slot m), rows 8-15 in
      // lanes 16-31 (slot m-8)
      int base = (lane < 8) ? 0 : 16;
      int slot = (lane < 8) ? lane : (lane - 8);
      float ssum = 0.0f;
      for (int i = 0; i < 16; ++i) ssum += red[wave][base + i][slot];
      outv[tokBase + lane] = ssum;
    }
  }
}

// ---------------------------------------------------------------------------
// Kernel 2: per head (8 blocks x 256 threads).
//   c[p]   = max_t (g_k[t] + u[t,p])
//   M[r,p] = sum_t phi_k[t,r] * exp(g_k[t]+u[t,p]-c[p])      (WMMA, K=512)
//   y[s,p] = f_q[s] + c[p] + log( sum_r phi_q[s,r]*M[r,p] ) - log S
// ---------------------------------------------------------------------------
__global__ __launch_bounds__(256) void cs_attn_kernel(
    const float* __restrict__ phi_q, const float* __restrict__ phi_kT,
    const float* __restrict__ u_ws, const float* __restrict__ f_q,
    const float* __restrict__ g_k, float* __restrict__ out) {
  __shared__ float ET[32 * 260];  // E^T half-tile: [p][t_local]
  __shared__ float MT[32 * 34];   // M^T: [p][r]
  __shared__ float pmax[8][32];
  __shared__ float c_s[32];

  const int h = blockIdx.x;
  const int tid = threadIdx.x;
  const int wave = tid >> 5;
  const int lane = tid & 31;
  const int lh = lane & 15;
  const int kh = lane >> 4;

  // Prefetch this head's phi tiles while we do the max-reduction.
  __builtin_prefetch(phi_kT + h * 16384 + tid * 64, 0, 1);
  __builtin_prefetch(phi_q + h * 16384 + tid * 64, 0, 1);

  // ---- c[p] ----
  {
    int p = tid & 31, chunk = tid >> 5;
    float m = -3.4e38f;
    for (int j = 0; j < 64; ++j) {
      int t = chunk * 64 + j;
      m = fmaxf(m, g_k[h * 512 + t] + u_ws[(h * 512 + t) * 32 + p]);
    }
    pmax[chunk][p] = m;
  }
  __syncthreads();
  if (tid < 32) {
    float m = pmax[0][tid];
    for (int c = 1; c < 8; ++c) m = fmaxf(m, pmax[c][tid]);
    c_s[tid] = m;
  }
  __syncthreads();

  // ---- M accumulation: waves 0..3 own the four 16x16 tiles of M ----
  const int mt = wave >> 1, ntw = wave & 1;
  v8f macc = {};
  for (int half = 0; half < 2; ++half) {
    int p = tid >> 3, tc = tid & 7;
    for (int j = 0; j < 32; ++j) {
      int tl = tc * 32 + j;
      int t = half * 256 + tl;
      ET[p * 260 + tl] =
          expf(g_k[h * 512 + t] + u_ws[(h * 512 + t) * 32 + p] - c_s[p]);
    }
    __syncthreads();
    if (wave < 4) {
      for (int kk = 0; kk < 64; ++kk) {
        int k0 = kk * 4 + 2 * kh;
        v2f a = *(const v2f*)&phi_kT[(h * 32 + mt * 16 + lh) * 512 +
                                     half * 256 + k0];
        v2f b = *(const v2f*)&ET[(ntw * 16 + lh) * 260 + k0];
        macc = wmma4(a, b, macc);
      }
    }
    __syncthreads();
  }
  if (wave < 4) {
    for (int j = 0; j < 8; ++j) {
      int r = mt * 16 + j + 8 * kh;
      int p = ntw * 16 + lh;
      MT[p * 34 + r] = macc[j];  // store transposed for the final GEMM
    }
  }
  __syncthreads();

  // ---- final: y = f_q + c + log(phi_q @ M) - log S ----
  const float logS = logf(512.0f);
  for (int it = 0; it < 8; ++it) {
    int tileId = wave * 8 + it;
    int st = tileId >> 1, nt2 = tileId & 1;
    v8f acc = {};
    for (int kk = 0; kk < 8; ++kk) {
      int r0 = kk * 4 + 2 * kh;
      v2f a = *(const v2f*)&phi_q[(h * 512 + st * 16 + lh) * 32 + r0];
      v2f b = *(const v2f*)&MT[(nt2 * 16 + lh) * 34 + r0];
      acc = wmma4(a, b, acc);
    }
    for (int j = 0; j < 8; ++j) {
      int s = st * 16 + j + 8 * kh;
      int p = nt2 * 16 + lh;
      out[(h * 512 + s) * 32 + p] =
          f_q[h * 512 + s] + c_s[p] + logf(acc[j]) - logS;
    }
  }
}

extern "C" void kernel_launch(void* const* d_in, const int* in_sizes, int n_in,
                              void* d_out, int out_size, void* d_ws,
                              size_t ws_size, hipStream_t stream) {
  const float* q = (const float*)d_in[0];
  const float* k = (const float*)d_in[1];
  const float* v = (const float*)d_in[2];
  const float* sq_raw1 = (const float*)d_in[3];
  const float* sq_b1 = (const float*)d_in[4];
  const float* sq_raw2 = (const float*)d_in[5];
  const float* sq_b2 = (const float*)d_in[6];
  const float* sk_raw1 = (const float*)d_in[7];
  const float* sk_b1 = (const float*)d_in[8];
  const float* sk_raw2 = (const float*)d_in[9];
  const float* sk_b2 = (const float*)d_in[10];
  const float* Wh = (const float*)d_in[11];
  const float* Wv = (const float*)d_in[12];

  float* ws = (float*)d_ws;
  float* phi_q = ws;                // [h][s][p]  8*512*32 = 131072
  float* phi_kT = ws + 131072;      // [h][r][t]  131072
  float* u_ws = ws + 262144;        // [h][t][p]  131072
  float* f_q = ws + 393216;         // 4096
  float* g_k = ws + 397312;         // 4096

  cs_token_kernel<<<64, 128, 0, stream>>>(q, k, v, sq_raw1, sq_b1, sq_raw2,
                                          sq_b2, sk_raw1, sk_b1, sk_raw2,
                                          sk_b2, Wh, Wv, phi_q, phi_kT, u_ws,
                                          f_q, g_k);
  cs_attn_kernel<<<8, 256, 0, stream>>>(phi_q, phi_kT, u_ws, f_q, g_k,
                                        (float*)d_out);
}